// HyperConnection_40432822124578
// MI455X (gfx1250) — compile-verified
//
#include <hip/hip_runtime.h>
#include <hip/hip_bf16.h>
#include <math.h>

typedef __attribute__((ext_vector_type(2))) float v2f;
typedef __attribute__((ext_vector_type(8))) float v8f;

#define BATCH_PER_WG 16
#define ND 4096          // n*D = 4*1024
#define DD 1024          // D
#define SINK_ITERS 30
#define EPS 1e-8f

// One workgroup (256 threads = 8 wave32) handles 16 batch rows.
__global__ __launch_bounds__(256) void hyperconn_fused_kernel(
    const float* __restrict__ x,     // (B, 4, 1024) row-major => (B, 4096)
    const float* __restrict__ Wq,    // (16, 4096)
    const float* __restrict__ Wb,    // (16,)
    const float* __restrict__ bq,    // (4,4) flat = (16,)
    float* __restrict__ y)           // (B, 4, 1024)
{
    __shared__ float hpart[8 * 256]; // per-wave 16x16 partial accumulators
    __shared__ float hmat[256];      // final 16x16 H (rows = batch, cols = 4x4 entries)

    const int tid  = threadIdx.x;
    const int wave = tid >> 5;
    const int lane = tid & 31;
    const int b0   = blockIdx.x * BATCH_PER_WG;

    // ---------------- Phase 1: H_tilde = x @ W.T via V_WMMA_F32_16X16X4_F32 --------
    // A (16x4, f32): lanes 0-15 -> M=0..15, K={k,k+1}; lanes 16-31 -> M=0..15, K={k+2,k+3}
    // B (4x16, f32): lanes 0-15 -> N=0..15, K={k,k+1}; lanes 16-31 -> N=0..15, K={k+2,k+3}
    const int r  = lane & 15;         // batch row for A / output col for B
    const int kk = (lane >> 4) << 1;  // K sub-offset {0,2}
    const float* aptr = x  + (size_t)(b0 + r) * ND + kk;
    const float* bptr = Wq + (size_t)r * ND + kk;   // B[k][n] = Wq[n][k]

    v8f acc = {};
    const int kbeg = wave * (ND / 8);
    const int kend = kbeg + (ND / 8);
    #pragma unroll 4
    for (int k = kbeg; k < kend; k += 4) {
        v2f a = *(const v2f*)(aptr + k);
        v2f b = *(const v2f*)(bptr + k);
        // 8-arg form: (neg_a, A, neg_b, B, c_mod, C, reuse_a, reuse_b)
        acc = __builtin_amdgcn_wmma_f32_16x16x4_f32(
            false, a, false, b, (short)0, acc, false, false);
    }

    // D layout: VGPR v holds M = v + 8*(lane>>4), N = lane&15
    {
        const int mhi = (lane >> 4) * 8;
        #pragma unroll
        for (int v = 0; v < 8; ++v)
            hpart[wave * 256 + (mhi + v) * 16 + r] = acc[v];
    }
    __syncthreads();

    // Cross-wave reduction + bias + exp. tid = M*16 + N of the 16x16 tile.
    {
        float s = 0.0f;
        #pragma unroll
        for (int w = 0; w < 8; ++w) s += hpart[w * 256 + tid];
        const int col = tid & 15;
        s += Wb[col] + bq[col];
        hmat[tid] = expf(s);
    }
    __syncthreads();

    // ---------------- Phase 2: Sinkhorn, one lane per batch row (all registers) ----
    if (tid < 16) {
        float m[16];
        #pragma unroll
        for (int c = 0; c < 16; ++c) m[c] = hmat[tid * 16 + c];
        for (int it = 0; it < SINK_ITERS; ++it) {
            #pragma unroll
            for (int i = 0; i < 4; ++i) {   // row normalize
                float s  = m[4*i] + m[4*i+1] + m[4*i+2] + m[4*i+3];
                float rs = 1.0f / (s + EPS);
                m[4*i] *= rs; m[4*i+1] *= rs; m[4*i+2] *= rs; m[4*i+3] *= rs;
            }
            #pragma unroll
            for (int j = 0; j < 4; ++j) {   // col normalize
                float s  = m[j] + m[j+4] + m[j+8] + m[j+12];
                float rs = 1.0f / (s + EPS);
                m[j] *= rs; m[j+4] *= rs; m[j+8] *= rs; m[j+12] *= rs;
            }
        }
        #pragma unroll
        for (int c = 0; c < 16; ++c) hmat[tid * 16 + c] = m[c];
    }
    __syncthreads();

    // ---------------- Phase 3: y[b] = H[b] (4x4) @ x[b] (4x1024), float4-vectorized
    {
        const int bl  = tid >> 4;    // local batch row 0..15
        const int l16 = tid & 15;
        float h[16];
        #pragma unroll
        for (int c = 0; c < 16; ++c) h[c] = hmat[bl * 16 + c];

        const float4* xb = (const float4*)(x + (size_t)(b0 + bl) * ND);
        float4*       yb = (float4*)(y + (size_t)(b0 + bl) * ND);

        #pragma unroll 2
        for (int dd = l16; dd < DD / 4; dd += 16) {
            float4 x0 = xb[0 * 256 + dd];
            float4 x1 = xb[1 * 256 + dd];
            float4 x2 = xb[2 * 256 + dd];
            float4 x3 = xb[3 * 256 + dd];
            #pragma unroll
            for (int i = 0; i < 4; ++i) {
                float4 o;
                o.x = h[4*i]*x0.x + h[4*i+1]*x1.x + h[4*i+2]*x2.x + h[4*i+3]*x3.x;
                o.y = h[4*i]*x0.y + h[4*i+1]*x1.y + h[4*i+2]*x2.y + h[4*i+3]*x3.y;
                o.z = h[4*i]*x0.z + h[4*i+1]*x1.z + h[4*i+2]*x2.z + h[4*i+3]*x3.z;
                o.w = h[4*i]*x0.w + h[4*i+1]*x1.w + h[4*i+2]*x2.w + h[4*i+3]*x3.w;
                yb[i * 256 + dd] = o;
            }
        }
    }
}

extern "C" void kernel_launch(void* const* d_in, const int* in_sizes, int n_in,
                              void* d_out, int out_size, void* d_ws, size_t ws_size,
                              hipStream_t stream) {
    const float* x  = (const float*)d_in[0];   // (B,4,1024)
    const float* Wq = (const float*)d_in[1];   // (16,4096)
    const float* Wb = (const float*)d_in[2];   // (16,)
    const float* bq = (const float*)d_in[3];   // (4,4)
    float* y = (float*)d_out;

    const int Btot = in_sizes[0] / ND;         // 32768
    const int grid = Btot / BATCH_PER_WG;      // 2048
    hipLaunchKernelGGL(hyperconn_fused_kernel, dim3(grid), dim3(256), 0, stream,
                       x, Wq, Wb, bq, y);
}